// RegressionLoss_26302379720779
// MI455X (gfx1250) — compile-verified
//
#include <hip/hip_runtime.h>
#include <hip/hip_bf16.h>
#include <stdint.h>

#define B_ 8
#define H_ 320
#define W_ 320
#define T_ 16
#define HW_ (H_ * W_)
#define NTOT_ (T_ * HW_)
#define HALF_NTOT_ (NTOT_ / 2)
#define HARD_NUM 128
#define RAND_NUM 128
#define THETA_COEF_ 10.0f

typedef __attribute__((ext_vector_type(2))) float v2f;
typedef __attribute__((ext_vector_type(8))) float v8f;

// ---------------- Threefry-2x32 (matches jax.random) ----------------
__device__ __forceinline__ void threefry2x32(uint32_t k0, uint32_t k1,
                                             uint32_t x0, uint32_t x1,
                                             uint32_t& o0, uint32_t& o1) {
  const uint32_t ks2 = k0 ^ k1 ^ 0x1BD11BDAu;
  uint32_t ks[3] = {k0, k1, ks2};
  x0 += ks[0]; x1 += ks[1];
  const int rotA[4] = {13, 15, 26, 6};
  const int rotB[4] = {17, 29, 16, 24};
  #pragma unroll
  for (int blk = 0; blk < 5; ++blk) {
    #pragma unroll
    for (int j = 0; j < 4; ++j) {
      int r = (blk & 1) ? rotB[j] : rotA[j];
      x0 += x1;
      x1 = (x1 << r) | (x1 >> (32 - r));
      x1 ^= x0;
    }
    x0 += ks[(blk + 1) % 3];
    x1 += ks[(blk + 2) % 3] + (uint32_t)(blk + 1);
  }
  o0 = x0; o1 = x1;
}

// element j of jax.random.split(key(42), 8) flattened to 16 u32s
__device__ __forceinline__ uint32_t split_elem42(int j) {
  uint32_t o0, o1;
  if (j < 8) { threefry2x32(0u, 42u, (uint32_t)j, (uint32_t)(j + 8), o0, o1); return o0; }
  threefry2x32(0u, 42u, (uint32_t)(j - 8), (uint32_t)j, o0, o1); return o1;
}

__device__ __forceinline__ void image_rng_key(int b, uint32_t& k0, uint32_t& k1) {
  k0 = split_elem42(2 * b);
  k1 = split_elem42(2 * b + 1);
}

// jax.random.uniform(rng, (NTOT,))[f]
__device__ __forceinline__ float uniform_r(uint32_t k0, uint32_t k1, uint32_t f) {
  uint32_t o0, o1, bits;
  if (f < (uint32_t)HALF_NTOT_) { threefry2x32(k0, k1, f, f + HALF_NTOT_, o0, o1); bits = o0; }
  else { threefry2x32(k0, k1, f - HALF_NTOT_, f, o0, o1); bits = o1; }
  return __uint_as_float((bits >> 9) | 0x3f800000u) - 1.0f;
}

// ---------------- K0: zero per-image candidate counters ----------------
__global__ void k_zero(uint32_t* counters) {
  if (threadIdx.x < B_) counters[threadIdx.x] = 0;
}

// ---------------- K1: WMMA inside-test + candidate append ----------------
// grid (100, B), block 256 (8 waves). Each wave: 8 groups of 16 pixels.
// WMMA: A = [16 px × (px,py,1,0)], B-tile = edge line coeffs for 16 (t,e) cols.
__global__ __launch_bounds__(256) void k_scan(
    const float* __restrict__ confs,   // B,H,W,1
    const float* __restrict__ rects,   // B,T,12 (normalized)
    uint32_t* __restrict__ counters,
    uint64_t* __restrict__ confkeys,
    uint64_t* __restrict__ rkeys,
    int cap) {
  const int b = blockIdx.y;
  const int lane = threadIdx.x & 31;
  const int wave = threadIdx.x >> 5;
  const int waveGlobal = blockIdx.x * 8 + wave;  // 0..799

  uint32_t rk0, rk1;
  image_rng_key(b, rk0, rk1);

  // Build 4 B-operand tiles (invariant over pixels). Column n = 4*t_local + e.
  v2f btile[4];
  {
    const int n = lane & 15;
    const int tl = n >> 2, e = n & 3, e1 = (e + 1) & 3;
    #pragma unroll
    for (int tt = 0; tt < 4; ++tt) {
      const int tq = tt * 4 + tl;
      const float* q = rects + ((size_t)b * T_ + tq) * 12 + 4;
      const float x0 = q[2 * e] * (float)W_, y0 = q[2 * e + 1] * (float)H_;
      const float x1 = q[2 * e1] * (float)W_, y1 = q[2 * e1 + 1] * (float)H_;
      const float ex = x1 - x0, ey = y1 - y0;
      v2f bb;
      if (lane < 16) { bb.x = -ey;            bb.y = ex;   }  // K=0,1 rows
      else           { bb.x = ey*x0 - ex*y0;  bb.y = 0.0f; }  // K=2,3 rows
      btile[tt] = bb;
    }
  }

  const int myT = lane & 15;        // t handled by this lane in sign phase
  const int myTile = myT >> 2;
  const int myShiftBase = (myT & 3) * 4;

  uint32_t* ctr = counters + b;
  uint64_t* ck = confkeys + (size_t)b * cap;
  uint64_t* rk = rkeys + (size_t)b * cap;
  const float* confb = confs + (size_t)b * HW_;

  for (int g = 0; g < 8; ++g) {
    const int pixBase = (waveGlobal * 8 + g) * 16;
    __builtin_prefetch(confb + pixBase, 0, 0);  // global_prefetch_b8

    v2f a;
    if (lane < 16) {
      const int p = pixBase + lane;
      a.x = (float)(p % W_) + 0.5f;
      a.y = (float)(p / W_) + 0.5f;
    } else { a.x = 1.0f; a.y = 0.0f; }

    uint32_t pos8[8], neg8[8];
    #pragma unroll
    for (int tt = 0; tt < 4; ++tt) {
      v8f c = {};
      v8f d = __builtin_amdgcn_wmma_f32_16x16x4_f32(
          false, a, false, btile[tt], (short)0, c, false, false);
      #pragma unroll
      for (int v = 0; v < 8; ++v) {
        const uint32_t pm = (uint32_t)__ballot(d[v] >= 0.0f);
        const uint32_t nm = (uint32_t)__ballot(d[v] <= 0.0f);
        if (tt == myTile) { pos8[v] = pm; neg8[v] = nm; }
      }
    }

    // lane L handles (m = 2*i + L/16, t = L%16) -> all 256 (m,t) pairs
    #pragma unroll
    for (int i = 0; i < 8; ++i) {
      const int m = 2 * i + (lane >> 4);
      const int v = m & 7;
      const int shift = ((m >> 3) << 4) + myShiftBase;
      const bool inside = (((pos8[v] >> shift) & 0xF) == 0xF) ||
                          (((neg8[v] >> shift) & 0xF) == 0xF);
      const uint32_t mask = (uint32_t)__ballot(inside);
      const int cnt = __popc(mask);
      if (cnt) {
        uint32_t base = 0;
        if (lane == 0) base = atomicAdd(ctr, (uint32_t)cnt);
        base = __shfl(base, 0, 32);  // wave-aggregated append
        if (inside) {
          const int p = pixBase + m;
          const uint32_t flat = (uint32_t)myT * HW_ + (uint32_t)p;  // t-major
          const uint32_t slot = base + __popc(mask & ((1u << lane) - 1u));
          if (slot < (uint32_t)cap) {
            const float conf = confb[p];
            ck[slot] = ((uint64_t)__float_as_uint(conf) << 21) | flat;
            const float r = uniform_r(rk0, rk1, flat);
            rk[slot] = ((uint64_t)__float_as_uint(r) << 21) | flat;
          }
        }
      }
    }
  }
}

// ---------------- K2: exact top-128 x2 selection + finalize ----------------
#define K2_THREADS 1024

__global__ __launch_bounds__(K2_THREADS) void k_select(
    const float* __restrict__ boxes,   // B,H,W,4 (pixels)
    const float* __restrict__ rects,   // B,T,12 (normalized)
    const float* __restrict__ thetas,  // B,H,W,1
    const float* __restrict__ tth,     // B,T,1
    const uint32_t* __restrict__ counters,
    uint64_t* __restrict__ confkeys,
    uint64_t* __restrict__ rkeys,
    int cap, float* __restrict__ out) {
  const int b = blockIdx.x;
  const int t = threadIdx.x;
  uint64_t* ck = confkeys + (size_t)b * cap;
  uint64_t* rk = rkeys + (size_t)b * cap;
  const int Nc = min((int)counters[b], cap);

  __shared__ uint64_t skey[K2_THREADS];
  __shared__ int sthr[K2_THREADS];
  __shared__ uint32_t selFlat[HARD_NUM + RAND_NUM];
  __shared__ int s_nsel;

  const uint64_t KMAX = ~0ull;
  const int ch = (Nc + K2_THREADS - 1) / K2_THREADS;
  const int beg = t * ch;
  const int end = min(beg + ch, Nc);

  if (t == 0) s_nsel = 0;

  // ---- phase A: 128 smallest conf keys (hard positives) ----
  uint64_t myKey = KMAX; int myPos = -1;
  for (int i = beg; i < end; ++i) { uint64_t k = ck[i]; if (k < myKey) { myKey = k; myPos = i; } }
  __syncthreads();
  for (int it = 0; it < HARD_NUM; ++it) {
    skey[t] = myKey; sthr[t] = t;
    __syncthreads();
    for (int off = K2_THREADS / 2; off > 0; off >>= 1) {
      if (t < off && skey[t + off] < skey[t]) { skey[t] = skey[t + off]; sthr[t] = sthr[t + off]; }
      __syncthreads();
    }
    if (skey[0] == KMAX) break;  // uniform
    if (t == sthr[0]) {
      selFlat[it] = (uint32_t)(myKey & 0x1FFFFFu);
      ck[myPos] = KMAX;
      rk[myPos] = KMAX;  // hard elements excluded from random phase
      myKey = KMAX; myPos = -1;
      for (int i = beg; i < end; ++i) { uint64_t k = ck[i]; if (k < myKey) { myKey = k; myPos = i; } }
    }
    if (t == 0) s_nsel = it + 1;
    __syncthreads();
  }
  __syncthreads();
  const int nhard = s_nsel;

  // ---- phase B: 128 smallest threefry keys among remaining ----
  myKey = KMAX; myPos = -1;
  for (int i = beg; i < end; ++i) { uint64_t k = rk[i]; if (k < myKey) { myKey = k; myPos = i; } }
  __syncthreads();
  for (int it = 0; it < RAND_NUM; ++it) {
    skey[t] = myKey; sthr[t] = t;
    __syncthreads();
    for (int off = K2_THREADS / 2; off > 0; off >>= 1) {
      if (t < off && skey[t + off] < skey[t]) { skey[t] = skey[t + off]; sthr[t] = sthr[t + off]; }
      __syncthreads();
    }
    if (skey[0] == KMAX) break;
    if (t == sthr[0]) {
      selFlat[nhard + it] = (uint32_t)(myKey & 0x1FFFFFu);
      rk[myPos] = KMAX;
      myKey = KMAX; myPos = -1;
      for (int i = beg; i < end; ++i) { uint64_t k = rk[i]; if (k < myKey) { myKey = k; myPos = i; } }
    }
    if (t == 0) s_nsel = nhard + it + 1;
    __syncthreads();
  }
  __syncthreads();
  const int nsel = s_nsel;

  // ---- phase C: loss at selected elements only ----
  float contrib = 0.0f;
  if (t < nsel) {
    const uint32_t flat = selFlat[t];
    const int tq = (int)(flat / HW_);
    const int p = (int)(flat % HW_);
    const float* pb = boxes + ((size_t)b * HW_ + p) * 4;
    const float* tr = rects + ((size_t)b * T_ + tq) * 12;
    const float tb0 = tr[0] * (float)W_, tb1 = tr[1] * (float)H_;
    const float tb2 = tr[2] * (float)W_, tb3 = tr[3] * (float)H_;
    const float ix0 = fmaxf(pb[0], tb0), iy0 = fmaxf(pb[1], tb1);
    const float ix1 = fminf(pb[2], tb2), iy1 = fminf(pb[3], tb3);
    const float inter = fmaxf(ix1 - ix0, 0.0f) * fmaxf(iy1 - iy0, 0.0f);
    const float ap = fmaxf(pb[2] - pb[0], 0.0f) * fmaxf(pb[3] - pb[1], 0.0f);
    const float at = (tb2 - tb0) * (tb3 - tb1);
    const float uni = ap + at - inter;
    const float loc = -logf((inter + 1.0f) / (uni + 1.0f));
    const float ori = 1.0f - cosf(thetas[(size_t)b * HW_ + p] - tth[(size_t)b * T_ + tq]);
    contrib = loc + THETA_COEF_ * ori;
  }
  __shared__ float ssum[K2_THREADS];
  ssum[t] = contrib;
  __syncthreads();
  for (int off = K2_THREADS / 2; off > 0; off >>= 1) {
    if (t < off) ssum[t] += ssum[t + off];
    __syncthreads();
  }
  if (t == 0) out[b] = ssum[0] / (float)max(nsel, 1);
}

// ---------------- launch ----------------
extern "C" void kernel_launch(void* const* d_in, const int* in_sizes, int n_in,
                              void* d_out, int out_size, void* d_ws, size_t ws_size,
                              hipStream_t stream) {
  const float* confs  = (const float*)d_in[0];  // pred_confs  B,H,W,1
  const float* boxes  = (const float*)d_in[1];  // pred_boxes  B,H,W,4
  const float* rects  = (const float*)d_in[2];  // true_rects  B,T,12
  const float* thetas = (const float*)d_in[3];  // pred_thetas B,H,W,1
  const float* tth    = (const float*)d_in[4];  // true_thetas B,T,1
  float* out = (float*)d_out;

  uint32_t* counters = (uint32_t*)d_ws;
  const size_t avail = (ws_size > 256) ? (ws_size - 256) : 0;
  size_t capSz = avail / ((size_t)B_ * 16);  // 2 u64 keys per candidate
  if (capSz > 262144) capSz = 262144;        // worst-case valid count ~147k/image
  int cap = (int)capSz;
  if (cap < 1) cap = 1;
  uint64_t* confkeys = (uint64_t*)((char*)d_ws + 256);
  uint64_t* rkeys = confkeys + (size_t)B_ * cap;

  k_zero<<<1, 32, 0, stream>>>(counters);
  k_scan<<<dim3(100, B_), 256, 0, stream>>>(confs, rects, counters, confkeys, rkeys, cap);
  k_select<<<B_, K2_THREADS, 0, stream>>>(boxes, rects, thetas, tth, counters,
                                          confkeys, rkeys, cap, out);
}